// L1_42975442764623
// MI455X (gfx1250) — compile-verified
//
#include <hip/hip_runtime.h>

// Problem constants (from reference: B=32, S=3, T=1000, F=257)
#define NB     32
#define NS     3
#define NE     257000      // T*F
#define NE4    64250       // NE/4 (float4 groups)
#define NPAIR  9           // S*S pairwise sums
#define NPERM  6           // 3!
#define CHUNKS 25          // blocks per batch element in y
#define TPB    256         // 8 waves (wave32)

typedef __attribute__((ext_vector_type(2))) float v2f;
typedef __attribute__((ext_vector_type(8))) float v8f;

// ---------------------------------------------------------------------------
// Compile-time packed permutation structure (itertools.permutations(range(3))):
//   p0=(0,1,2) p1=(0,2,1) p2=(1,0,2) p3=(1,2,0) p4=(2,0,1) p5=(2,1,0)
//
// MPACK: for kk=j*3+k in 0..8, 6-bit mask over p of (perms[p][k]==j):
//   kk:  0   1   2   3   4   5   6   7   8
// mask:  3  20  40  12  33  18  48  10   5
#define MPACK  ( 3ULL        | (20ULL << 6)  | (40ULL << 12) \
               | (12ULL << 18)| (33ULL << 24)| (18ULL << 30) \
               | (48ULL << 36)| (10ULL << 42)| ( 5ULL << 48) )
// PERMPACK: perms[p][k] as 2-bit digits at bit (p*6 + k*2):
//   p: 36, 24, 33, 9, 18, 6
#define PERMPACK ( 36ULL | (24ULL << 6) | (33ULL << 12) \
                 | ( 9ULL << 18) | (18ULL << 24) | (6ULL << 30) )

__device__ __forceinline__ float mval(int kk, int p) {
    // M[kk][p] = (perms[p][k]==j)/3, zero-padded for kk in [9,12) -- pure ALU
    if (kk >= NPAIR) return 0.0f;
    return ((MPACK >> (6 * kk + p)) & 1ULL) ? (1.0f / 3.0f) : 0.0f;
}

// ---------------------------------------------------------------------------
// Kernel 0: zero the 32x9 pairwise accumulator workspace (harness poisons ws)
// ---------------------------------------------------------------------------
__global__ void zero_ws_kernel(float* __restrict__ ws) {
    int i = threadIdx.x;
    if (i < NB * NPAIR) ws[i] = 0.0f;
}

// ---------------------------------------------------------------------------
// Kernel 1: HBM-bound streamer. Each element of pred/tgt is read exactly once
// (optimal ~197 MB traffic -> ~8.5us at 23.3 TB/s). float4 = global_load_b128.
// ---------------------------------------------------------------------------
__device__ __forceinline__ float absdiff4(float4 a, float4 c) {
    float d0 = fabsf(a.x - c.x);
    float d1 = fabsf(a.y - c.y);
    float d2 = fabsf(a.z - c.z);
    float d3 = fabsf(a.w - c.w);
    return (d0 + d1) + (d2 + d3);
}

__global__ __launch_bounds__(TPB) void pairwise_l1_kernel(
    const float* __restrict__ pred, const float* __restrict__ tgt,
    float* __restrict__ ws) {
    const int b   = blockIdx.x;   // batch element
    const int tid = threadIdx.x;
    const size_t base = (size_t)b * NS * NE;
    const float4* __restrict__ p0 = (const float4*)(pred + base);
    const float4* __restrict__ p1 = (const float4*)(pred + base + (size_t)NE);
    const float4* __restrict__ p2 = (const float4*)(pred + base + (size_t)2 * NE);
    const float4* __restrict__ t0 = (const float4*)(tgt + base);
    const float4* __restrict__ t1 = (const float4*)(tgt + base + (size_t)NE);
    const float4* __restrict__ t2 = (const float4*)(tgt + base + (size_t)2 * NE);

    float acc[NPAIR];
#pragma unroll
    for (int j = 0; j < NPAIR; ++j) acc[j] = 0.0f;

    const int stride = CHUNKS * TPB;
    for (int i = blockIdx.y * TPB + tid; i < NE4; i += stride) {
        float4 a0 = p0[i], a1 = p1[i], a2 = p2[i];
        float4 c0 = t0[i], c1 = t1[i], c2 = t2[i];
        acc[0] += absdiff4(a0, c0);  // j=0,k=0
        acc[1] += absdiff4(a0, c1);
        acc[2] += absdiff4(a0, c2);
        acc[3] += absdiff4(a1, c0);
        acc[4] += absdiff4(a1, c1);
        acc[5] += absdiff4(a1, c2);
        acc[6] += absdiff4(a2, c0);
        acc[7] += absdiff4(a2, c1);
        acc[8] += absdiff4(a2, c2);  // j=2,k=2
    }

    // wave32 butterfly reduction, then one f32 atomic per wave per pair
#pragma unroll
    for (int j = 0; j < NPAIR; ++j) {
        float v = acc[j];
#pragma unroll
        for (int off = 16; off > 0; off >>= 1) v += __shfl_down(v, off, 32);
        if ((tid & 31) == 0) atomicAdd(&ws[b * NPAIR + j], v);
    }
}

// ---------------------------------------------------------------------------
// Kernel 2: finalize. loss_set[32,6] = PW[32,9] x M[9,6] as an EXACT f32 GEMM
// on the matrix core: V_WMMA_F32_16X16X4_F32, 2 M-tiles x 3 K-chunks
// (K padded 9->12). B-matrix built from packed immediates (no memory, no
// divergence). Then per-b min/argmin (first-min, matching jnp.argmin) and the
// mean over b. Single wave32 block; EXEC all-ones at every WMMA.
// ---------------------------------------------------------------------------
__global__ __launch_bounds__(32) void finalize_kernel(
    const float* __restrict__ ws, float* __restrict__ out) {
    __shared__ float pw[NB][12];   // K padded to 12 (3 chunks of 4)
    __shared__ float ls[NB][16];   // loss_set, cols 6..15 unused

    const int lane = threadIdx.x;  // 0..31, one wave
    const int n    = lane & 15;
    const int half = lane >> 4;

    const float invE = 1.0f / (float)NE;  // pairwise mean; final /3 lives in M
#pragma unroll
    for (int i = 0; i < 12; ++i)
        pw[lane][i] = (i < NPAIR) ? ws[lane * NPAIR + i] * invE : 0.0f;
    __syncthreads();

    // A 16x4 f32 layout: lanes 0-15 -> M=lane, VGPRs = K kc+0,kc+1;
    //                    lanes 16-31 -> M=lane-16, VGPRs = K kc+2,kc+3.
    // B 4x16 f32 layout: VGPR0 = row kc+(half?2:0), VGPR1 = row kc+(half?3:1),
    //                    N = lane%16.  C/D: VGPR r -> M = r + 8*half, N = lane%16.
#pragma unroll
    for (int tile = 0; tile < 2; ++tile) {
        v8f c = {};
#pragma unroll
        for (int kc3 = 0; kc3 < 3; ++kc3) {
            const int kc  = kc3 * 4;
            const int row = tile * 16 + n;
            const int ka  = kc + 2 * half;
            v2f a, bb;
            a.x  = pw[row][ka];
            a.y  = pw[row][ka + 1];
            bb.x = mval(kc + (half ? 2 : 0), n & 7);  // cols n>=6 hold junk (n&7 wraps), never read
            bb.y = mval(kc + (half ? 3 : 1), n & 7);
            // D = A x B + C  (exact f32 matrix op)
            c = __builtin_amdgcn_wmma_f32_16x16x4_f32(
                /*neg_a=*/false, a, /*neg_b=*/false, bb,
                /*c_mod=*/(short)0, c, /*reuse_a=*/false, /*reuse_b=*/false);
        }
#pragma unroll
        for (int r = 0; r < 8; ++r) {
            int m = tile * 16 + r + (half ? 8 : 0);
            ls[m][n] = c[r];
        }
    }
    __syncthreads();

    // per-batch argmin over 6 permutations (strict < keeps first minimum)
    float mn = ls[lane][0];
    int arg = 0;
#pragma unroll
    for (int p = 1; p < NPERM; ++p) {
        float v = ls[lane][p];
        if (v < mn) { mn = v; arg = p; }
    }
    const unsigned long long pd = PERMPACK >> (arg * 6);
    out[1 + lane * NS + 0] = (float)((int)(pd      ) & 3);
    out[1 + lane * NS + 1] = (float)((int)(pd >> 2 ) & 3);
    out[1 + lane * NS + 2] = (float)((int)(pd >> 4 ) & 3);

    // mean of per-batch min loss
    float s = mn;
#pragma unroll
    for (int off = 16; off > 0; off >>= 1) s += __shfl_down(s, off, 32);
    if (lane == 0) out[0] = s * (1.0f / (float)NB);
}

// ---------------------------------------------------------------------------
extern "C" void kernel_launch(void* const* d_in, const int* in_sizes, int n_in,
                              void* d_out, int out_size, void* d_ws, size_t ws_size,
                              hipStream_t stream) {
    (void)in_sizes; (void)n_in; (void)out_size; (void)ws_size;
    const float* pred = (const float*)d_in[0];  // predicted [32,3,1000,257] f32
    const float* tgt  = (const float*)d_in[1];  // padded_srcs_feats, same shape
    float* out = (float*)d_out;                 // [0]=mean min loss, [1..96]=best perms
    float* ws  = (float*)d_ws;                  // 32*9 f32 pairwise accumulators

    zero_ws_kernel<<<1, 288, 0, stream>>>(ws);
    dim3 grid(NB, CHUNKS);
    pairwise_l1_kernel<<<grid, TPB, 0, stream>>>(pred, tgt, ws);
    finalize_kernel<<<1, 32, 0, stream>>>(ws, out);
}